// SoftRasterizer_19920058319516
// MI455X (gfx1250) — compile-verified
//
#include <hip/hip_runtime.h>
#include <math.h>

// ---------------- problem constants ----------------
#define ISZ       64
#define NFACE     1024
#define NBATCH    4
#define EPSV      0.001f
#define INV_GAMMA 1.0e4f      // 1/GAMMA
#define INV_SIGMA 1.0e5f      // 1/SIGMA
#define THRESH    9.210240e-05f  // SIGMA * ln(1/DIST_EPS - 1)

typedef __attribute__((ext_vector_type(2))) float v2f;
typedef __attribute__((ext_vector_type(8))) float v8f;
typedef int v4i_t __attribute__((vector_size(16)));  // matches async-LDS builtin param

// ---- CDNA5 feature probes (device pass only; host pass takes fallbacks) ----
#if defined(__HIP_DEVICE_COMPILE__) && __has_builtin(__builtin_amdgcn_wmma_f32_16x16x4_f32)
#define HAVE_WMMA4 1
#else
#define HAVE_WMMA4 0
#endif

#if defined(__HIP_DEVICE_COMPILE__) && __has_builtin(__builtin_amdgcn_global_load_async_to_lds_b128)
#define HAVE_ASYNC 1
#else
#define HAVE_ASYNC 0
#endif

__device__ __forceinline__ float satf(float x) { return fminf(fmaxf(x, 0.f), 1.f); }

// fast hardware reciprocal (v_rcp_f32, ~1 ulp) instead of IEEE div sequence
__device__ __forceinline__ float frcp(float x) {
#if defined(__HIP_DEVICE_COMPILE__)
  return __builtin_amdgcn_rcpf(x);
#else
  return 1.0f / x;
#endif
}

// Copy one 12KB face-constant tile (128 faces * 24 floats) global -> LDS.
// 768 b128 chunks / 256 threads = 3 per thread.
__device__ __forceinline__ void copy_tile(float* dst_lds, const float* src_glb, int tid) {
#if HAVE_ASYNC
#pragma unroll
  for (int k = 0; k < 3; ++k) {
    int chunk = tid + k * 256;
    __builtin_amdgcn_global_load_async_to_lds_b128(
        (__attribute__((address_space(1))) v4i_t*)(src_glb + chunk * 4),
        (__attribute__((address_space(3))) v4i_t*)(dst_lds + chunk * 4),
        0, 0);
  }
#else
#pragma unroll
  for (int k = 0; k < 3; ++k) {
    int chunk = tid + k * 256;
    reinterpret_cast<float4*>(dst_lds)[chunk] =
        reinterpret_cast<const float4*>(src_glb)[chunk];
  }
#endif
}

template <int N>
__device__ __forceinline__ void wait_async() {
#if HAVE_ASYNC
#if __has_builtin(__builtin_amdgcn_s_wait_asynccnt)
  __builtin_amdgcn_s_wait_asynccnt(N);
#else
  asm volatile("s_wait_asynccnt %0" ::"i"(N) : "memory");
#endif
#endif
}

// ------------------------------------------------------------------
// Kernel 1: per-face constant precompute (4096 faces -> 24 floats each)
// layout: [a0 b0 c0 a1 b1 c1 | x0 y0 x1 y1 x2 y2 | iz0 iz1 iz2 | ft(9)]
// ------------------------------------------------------------------
__global__ __launch_bounds__(256) void sr_precompute(const float* __restrict__ fv,
                                                     const float* __restrict__ ft,
                                                     float* __restrict__ fcst) {
  int i = blockIdx.x * 256 + threadIdx.x;
  if (i >= NBATCH * NFACE) return;
  const float* v = fv + (size_t)i * 9;
  const float* t = ft + (size_t)i * 9;
  float x0 = v[0], y0 = v[1], z0 = v[2];
  float x1 = v[3], y1 = v[4], z1 = v[5];
  float x2 = v[6], y2 = v[7], z2 = v[8];
  float det = (y1 - y2) * (x0 - x2) + (x2 - x1) * (y0 - y2);
  if (fabsf(det) < 1e-10f) det = 1e-10f;
  float inv = frcp(det);
  float A0 = (y1 - y2) * inv, B0 = (x2 - x1) * inv;
  float C0 = -A0 * x2 - B0 * y2;
  float A1 = (y2 - y0) * inv, B1 = (x0 - x2) * inv;
  float C1 = -A1 * x2 - B1 * y2;
  float* o = fcst + (size_t)i * 24;
  o[0] = A0; o[1] = B0; o[2] = C0; o[3] = A1; o[4] = B1; o[5] = C1;
  o[6] = x0; o[7] = y0; o[8] = x1; o[9] = y1; o[10] = x2; o[11] = y2;
  o[12] = frcp(z0); o[13] = frcp(z1); o[14] = frcp(z2);
#pragma unroll
  for (int k = 0; k < 9; ++k) o[15 + k] = t[k];
}

// ------------------------------------------------------------------
// Kernel 2: main pass. Block = 256 thr (8 waves) = 16x8 pixel tile.
// Wave = one 16-pixel row; WMMA 16x16x4: 16 pixels x 16 faces barycentrics.
// Online-softmax accumulation over faces; writes D and znm (into aggr slot),
// images, and per-pixel (m, denom) for the finalize pass.
// ------------------------------------------------------------------
__global__ __launch_bounds__(256) void sr_main(const float* __restrict__ fcst,
                                               float* __restrict__ Dout,
                                               float* __restrict__ Zout,
                                               float* __restrict__ imgs,
                                               float* __restrict__ wsM,
                                               float* __restrict__ wsDen) {
  __shared__ __align__(16) float sh[2][128 * 24];  // 2 x 12KB double buffer

  const int tid = threadIdx.x;
  const int wave = tid >> 5;
  const int lane = tid & 31;
  const int hi = lane >> 4;            // 0 -> pixels 0..7, 1 -> pixels 8..15
  const int b = blockIdx.z;
  const int xbase = blockIdx.x * 16;
  const int y = blockIdx.y * 8 + wave;

  const float py = (63.0f - 2.0f * (float)y) * (1.0f / 64.0f);
  const float px0 = (2.0f * (float)(xbase + 8 * hi) + 1.0f - 64.0f) * (1.0f / 64.0f);

#if HAVE_WMMA4
  // A (16x4): lanes 0-15: M=lane, K=(0,1)=(px,py); lanes 16-31: K=(2,3)=(1,0)
  v2f avec;
  if (lane < 16) {
    avec[0] = (2.0f * (float)(xbase + lane) + 1.0f - 64.0f) * (1.0f / 64.0f);
    avec[1] = py;
  } else {
    avec[0] = 1.0f;
    avec[1] = 0.0f;
  }
#endif

  float m[8], dsum[8], r0[8], r1[8], r2[8], ap[8];
#pragma unroll
  for (int v = 0; v < 8; ++v) {
    m[v] = EPSV; dsum[v] = 0.f;
    r0[v] = 0.f; r1[v] = 0.f; r2[v] = 0.f;
    ap[v] = 1.f;
  }

  const float* gbase = fcst + (size_t)b * (NFACE * 24);
  copy_tile(&sh[0][0], gbase, tid);

  for (int tile = 0; tile < 8; ++tile) {
    const int cur = tile & 1;
    if (tile < 7) copy_tile(&sh[cur ^ 1][0], gbase + (size_t)(tile + 1) * 128 * 24, tid);
    if (tile < 7) wait_async<3>(); else wait_async<0>();
    __syncthreads();

    const float* shbuf = &sh[cur][0];
    for (int sub = 0; sub < 8; ++sub) {
      const int fl = sub * 16 + (lane & 15);
      const float* fc = shbuf + fl * 24;
      float a0 = fc[0], b0 = fc[1], c0 = fc[2];
      float a1 = fc[3], b1 = fc[4], c1 = fc[5];
      float x0 = fc[6], y0 = fc[7], x1 = fc[8], y1 = fc[9], x2 = fc[10], y2 = fc[11];
      float iz0 = fc[12], iz1 = fc[13], iz2 = fc[14];
      float t00 = fc[15], t01 = fc[16], t02 = fc[17];
      float t10 = fc[18], t11 = fc[19], t12 = fc[20];
      float t20 = fc[21], t21 = fc[22], t22 = fc[23];

      v8f w0v, w1v;
#if HAVE_WMMA4
      // B (4x16): lanes 0-15: K=(0,1)=(a,b); lanes 16-31: K=(2,3)=(c,0), N=lane%16
      v2f b0vec, b1vec;
      if (lane < 16) { b0vec[0] = a0; b0vec[1] = b0; b1vec[0] = a1; b1vec[1] = b1; }
      else           { b0vec[0] = c0; b0vec[1] = 0.f; b1vec[0] = c1; b1vec[1] = 0.f; }
      v8f cz = {0.f, 0.f, 0.f, 0.f, 0.f, 0.f, 0.f, 0.f};
      w0v = __builtin_amdgcn_wmma_f32_16x16x4_f32(false, avec, false, b0vec,
                                                  (short)0, cz, false, false);
      w1v = __builtin_amdgcn_wmma_f32_16x16x4_f32(false, avec, false, b1vec,
                                                  (short)0, cz, false, false);
#else
#pragma unroll
      for (int v = 0; v < 8; ++v) {
        float pxv = px0 + 0.03125f * (float)v;
        w0v[v] = a0 * pxv + b0 * py + c0;
        w1v[v] = a1 * pxv + b1 * py + c1;
      }
#endif
      float e0x = x1 - x0, e0y = y1 - y0; float il0 = frcp(e0x * e0x + e0y * e0y + 1e-12f);
      float e1x = x2 - x1, e1y = y2 - y1; float il1 = frcp(e1x * e1x + e1y * e1y + 1e-12f);
      float e2x = x0 - x2, e2y = y0 - y2; float il2 = frcp(e2x * e2x + e2y * e2y + 1e-12f);

      float dbuf[8], zbuf[8];
#pragma unroll
      for (int v = 0; v < 8; ++v) {
        float pxv = px0 + 0.03125f * (float)v;
        float w0 = w0v[v], w1 = w1v[v];
        float w2 = 1.f - w0 - w1;
        bool inside = (w0 > 0.f) && (w1 > 0.f) && (w2 > 0.f);
        float d2;
        { float qx = pxv - x0, qy = py - y0;
          float t = satf((qx * e0x + qy * e0y) * il0);
          float dx = qx - t * e0x, dy = qy - t * e0y; d2 = dx * dx + dy * dy; }
        { float qx = pxv - x1, qy = py - y1;
          float t = satf((qx * e1x + qy * e1y) * il1);
          float dx = qx - t * e1x, dy = qy - t * e1y; d2 = fminf(d2, dx * dx + dy * dy); }
        { float qx = pxv - x2, qy = py - y2;
          float t = satf((qx * e2x + qy * e2y) * il2);
          float dx = qx - t * e2x, dy = qy - t * e2y; d2 = fminf(d2, dx * dx + dy * dy); }
        float sg = inside ? 1.f : -1.f;
        float Dv = frcp(1.f + __expf(-sg * d2 * INV_SIGMA));
        if (!inside && d2 >= THRESH) Dv = 0.f;
        float wc0 = satf(w0), wc1 = satf(w1), wc2 = satf(w2);
        float is = frcp(wc0 + wc1 + wc2 + 1e-12f);
        float wn0 = wc0 * is, wn1 = wc1 * is, wn2 = wc2 * is;
        float zpinv = wn0 * iz0 + wn1 * iz1 + wn2 * iz2;
        float zp = frcp(zpinv);
        bool valid = (zp > 1.f) && (zp < 100.f);
        float coef = valid ? Dv : 0.f;
        float zn = (100.f - zp) * (1.f / 99.f);
        float znm = (coef > 0.f) ? zn : -1.f;
        dbuf[v] = Dv;
        zbuf[v] = znm;
        // online softmax update
        float mn = fmaxf(m[v], znm);
        float sc = __expf((m[v] - mn) * INV_GAMMA);
        float we = coef * __expf((znm - mn) * INV_GAMMA);
        m[v] = mn;
        dsum[v] = dsum[v] * sc + we;
        float rr = wn0 * t00 + wn1 * t10 + wn2 * t20;
        float gg = wn0 * t01 + wn1 * t11 + wn2 * t21;
        float bb = wn0 * t02 + wn1 * t12 + wn2 * t22;
        r0[v] = r0[v] * sc + we * rr;
        r1[v] = r1[v] * sc + we * gg;
        r2[v] = r2[v] * sc + we * bb;
        ap[v] *= (1.f - Dv);
      }
      int fglob = tile * 128 + fl;
      size_t off = (size_t)(b * NFACE + fglob) * 4096 + (size_t)y * 64 + xbase + 8 * hi;
      *reinterpret_cast<float4*>(Dout + off)     = make_float4(dbuf[0], dbuf[1], dbuf[2], dbuf[3]);
      *reinterpret_cast<float4*>(Dout + off + 4) = make_float4(dbuf[4], dbuf[5], dbuf[6], dbuf[7]);
      *reinterpret_cast<float4*>(Zout + off)     = make_float4(zbuf[0], zbuf[1], zbuf[2], zbuf[3]);
      *reinterpret_cast<float4*>(Zout + off + 4) = make_float4(zbuf[4], zbuf[5], zbuf[6], zbuf[7]);
    }
    __syncthreads();
  }

  // merge online states across the 16 lanes (one face column each) per pixel
#pragma unroll
  for (int stepv = 1; stepv < 16; stepv <<= 1) {
#pragma unroll
    for (int v = 0; v < 8; ++v) {
      float mo  = __shfl_xor(m[v],    stepv, 16);
      float don = __shfl_xor(dsum[v], stepv, 16);
      float ra  = __shfl_xor(r0[v],   stepv, 16);
      float ga  = __shfl_xor(r1[v],   stepv, 16);
      float ba  = __shfl_xor(r2[v],   stepv, 16);
      float apo = __shfl_xor(ap[v],   stepv, 16);
      float mn = fmaxf(m[v], mo);
      float sa = __expf((m[v] - mn) * INV_GAMMA);
      float sb = __expf((mo   - mn) * INV_GAMMA);
      m[v] = mn;
      dsum[v] = dsum[v] * sa + don * sb;
      r0[v] = r0[v] * sa + ra * sb;
      r1[v] = r1[v] * sa + ga * sb;
      r2[v] = r2[v] * sa + ba * sb;
      ap[v] *= apo;
    }
  }

  if ((lane & 15) == 0) {  // lanes 0 (pixels 0..7) and 16 (pixels 8..15)
#pragma unroll
    for (int v = 0; v < 8; ++v) {
      int xi = xbase + v + 8 * hi;
      float wbg = __expf((EPSV - m[v]) * INV_GAMMA);
      float dn = dsum[v] + wbg;
      float idn = frcp(dn);
      size_t po = (size_t)b * 4096 + (size_t)y * 64 + xi;
      imgs[(size_t)(b * 4 + 0) * 4096 + (size_t)y * 64 + xi] = r0[v] * idn;  // BG==0
      imgs[(size_t)(b * 4 + 1) * 4096 + (size_t)y * 64 + xi] = r1[v] * idn;
      imgs[(size_t)(b * 4 + 2) * 4096 + (size_t)y * 64 + xi] = r2[v] * idn;
      imgs[(size_t)(b * 4 + 3) * 4096 + (size_t)y * 64 + xi] = 1.f - ap[v];
      wsM[po] = m[v];
      wsDen[po] = dn;
    }
  }
}

// ------------------------------------------------------------------
// Kernel 3: aggr finalize (elementwise): aggr = w / denom,
// w = (znm>=0) ? D*exp((znm-m)/gamma) : 0   (znm was stashed in aggr slot)
// ------------------------------------------------------------------
__global__ __launch_bounds__(256) void sr_finalize(const float* __restrict__ Dout,
                                                   float* __restrict__ aggr,
                                                   const float* __restrict__ wsM,
                                                   const float* __restrict__ wsDen) {
  int idx = blockIdx.x * 256 + threadIdx.x;  // 0 .. 16777215
  int pix = idx & 4095;
  int b = idx >> 22;
  float mm = wsM[(b << 12) + pix];
  float idn = frcp(wsDen[(b << 12) + pix]);
  float z = aggr[idx];
  float Dv = Dout[idx];
  float w = (z >= 0.f) ? Dv * __expf((z - mm) * INV_GAMMA) : 0.f;
  aggr[idx] = w * idn;
}

// ------------------------------------------------------------------
extern "C" void kernel_launch(void* const* d_in, const int* in_sizes, int n_in,
                              void* d_out, int out_size, void* d_ws, size_t ws_size,
                              hipStream_t stream) {
  (void)in_sizes; (void)n_in; (void)out_size; (void)ws_size;
  const float* fv = (const float*)d_in[0];  // (4,1024,3,3)
  const float* ft = (const float*)d_in[1];  // (4,1024,3,3)
  float* out = (float*)d_out;
  float* imgs = out;                         // (4,4,64,64)
  float* Dout = out + 65536;                 // (4,1024,64,64)
  float* aggr = out + 65536 + 16777216;      // (4,1024,64,64)
  float* ws = (float*)d_ws;
  float* fcst  = ws;                         // 4096 faces * 24 floats
  float* wsM   = ws + 98304;                 // 4*64*64
  float* wsDen = ws + 98304 + 16384;         // 4*64*64

  sr_precompute<<<16, 256, 0, stream>>>(fv, ft, fcst);
  sr_main<<<dim3(4, 8, 4), 256, 0, stream>>>(fcst, Dout, aggr, imgs, wsM, wsDen);
  sr_finalize<<<65536, 256, 0, stream>>>(Dout, aggr, wsM, wsDen);
}